// trend_seasonality_decomposition_45629732553334
// MI455X (gfx1250) — compile-verified
//
#include <hip/hip_runtime.h>
#include <stdint.h>

#define L_LEN   8192
#define KW      25
#define HALF    12          // (K-1)/2
#define PHASES  24
#define NTERM   342         // 8208 / 24
#define LPAD    (L_LEN + 2*HALF)   // 8216
#define LPAD_A  8232        // allocated (padded, keeps slide read in-bounds)
#define CHUNK   33          // gcd(33,64)=1 -> conflict-free LDS banking
#define TPB     256
#define NROWS   (64*128)

typedef float        f4  __attribute__((ext_vector_type(4)));
typedef unsigned int v4u __attribute__((ext_vector_type(4)));
typedef int          v8i __attribute__((ext_vector_type(8)));
typedef int          v4i __attribute__((ext_vector_type(4)));

__device__ __forceinline__ void cswap(float& a, float& b) {
    float lo = fminf(a, b);
    float hi = fmaxf(a, b);
    a = lo; b = hi;
}

// full 5-element sorting network (Knuth, 9 comparators); median = middle element
__device__ __forceinline__ float median5(float a, float b, float c, float d, float e) {
    cswap(a, b); cswap(d, e); cswap(c, e); cswap(c, d); cswap(a, d);
    cswap(a, c); cswap(b, e); cswap(b, d); cswap(b, c);
    return c;
}

__global__ __launch_bounds__(TPB)
void decomp_kernel(const float* __restrict__ x, float* __restrict__ out)
{
    __shared__ float s_x[LPAD_A];        // padded series; x lives at [HALF, HALF+L)
    __shared__ float s_trend[L_LEN];
    __shared__ float s_pp[PHASES * 8];
    __shared__ float s_season[PHASES];

    const int tid = threadIdx.x;
    const int row = blockIdx.x;
    const float* xr = x + (size_t)row * L_LEN;

    // ---- Stage 1: DMA row into LDS via Tensor Data Mover (CDNA5 TDM) ----
#if defined(__HIP_DEVICE_COMPILE__) && __has_builtin(__builtin_amdgcn_tensor_load_to_lds) && __has_builtin(__builtin_amdgcn_s_wait_tensorcnt)
    if (tid < 32) {   // one wave issues the TDM op (EXEC-independent, per-wave)
        uint32_t lds_addr = (uint32_t)(uintptr_t)&s_x[HALF];   // LDS byte offset
        uint64_t ga = (uint64_t)(uintptr_t)xr;
        // D# group 0: count=1 | lds_addr | global_addr[56:0] | type=2
        v4u g0;
        g0[0] = 1u;                                      // count=1, user mode
        g0[1] = lds_addr;                                // lds_addr (bytes)
        g0[2] = (uint32_t)ga;                            // global_addr[31:0]
        g0[3] = (uint32_t)(ga >> 32) | (2u << 30);       // global_addr[56:32] | type=2
        // D# group 1: data_size=4B, tensor_dim0=tile_dim0=8192, dim1=1
        v8i g1;
        g1[0] = 0x00020000;      // workgroup_mask=0, data_size=2 (4 bytes)
        g1[1] = 0x20000000;      // tensor_dim0[15:0]=8192 in bits[31:16]
        g1[2] = 0x00010000;      // tensor_dim0[31:16]=0, tensor_dim1=1
        g1[3] = 0x20000000;      // tile_dim0=8192 in bits[31:16]
        g1[4] = 0x00000001;      // tile_dim1=1, tile_dim2=0
        g1[5] = 0x00002000;      // tensor_dim0_stride=8192 (low 32)
        g1[6] = 0x20000000;      // stride0 hi=0, tensor_dim1_stride[15:0]=8192
        g1[7] = 0;               // tensor_dim1_stride hi
        v4i gz4 = {0, 0, 0, 0};                  // groups 2/3 unused (<=2D tensor)
        v8i gz8 = {0, 0, 0, 0, 0, 0, 0, 0};      // extra descriptor words (unused)
        __builtin_amdgcn_tensor_load_to_lds(g0, g1, gz4, gz4, gz8, 0);
        __builtin_amdgcn_s_wait_tensorcnt(0);
    }
#else
    for (int i = tid; i < L_LEN / 4; i += TPB) {
        f4 v = ((const f4*)xr)[i];
        *(f4*)&s_x[HALF + i * 4] = v;
    }
#endif
    __syncthreads();

    // ---- Stage 2: median-of-5 edge padding ----
    if (tid == 0) {
        float fm = median5(s_x[HALF + 0], s_x[HALF + 1], s_x[HALF + 2],
                           s_x[HALF + 3], s_x[HALF + 4]);
        float em = median5(s_x[HALF + L_LEN - 5], s_x[HALF + L_LEN - 4],
                           s_x[HALF + L_LEN - 3], s_x[HALF + L_LEN - 2],
                           s_x[HALF + L_LEN - 1]);
        for (int i = 0; i < HALF; ++i) {
            s_x[i] = fm;
            s_x[HALF + L_LEN + i] = em;
        }
        for (int i = LPAD; i < LPAD_A; ++i) s_x[i] = 0.0f;  // slide overread guard
    }
    __syncthreads();

    // ---- Stage 3: 25-tap moving average, sliding-window, stride-33 chunks ----
    {
        int t0 = tid * CHUNK;
        if (t0 < L_LEN) {
            int tend = t0 + CHUNK; if (tend > L_LEN) tend = L_LEN;
            float s = 0.0f;
            #pragma unroll
            for (int j = 0; j < KW; ++j) s += s_x[t0 + j];
            for (int t = t0; t < tend; ++t) {
                s_trend[t] = s * (1.0f / 25.0f);
                s += s_x[t + KW] - s_x[t];   // slide (guarded by alloc padding)
            }
        }
    }
    __syncthreads();

    // ---- Stage 4: per-phase mean of residual over padded-to-8208 series ----
    // xpad = concat(res[0:8192], res[8168:8184]); seasonal[p] = mean of 342 terms
    if (tid < PHASES * 8) {
        int p  = tid % PHASES;
        int sg = tid / PHASES;
        float acc = 0.0f;
        for (int k = sg; k < NTERM; k += 8) {
            int pos = k * PHASES + p;
            int src = (pos < L_LEN) ? pos : (pos - PHASES);   // appended slice
            acc += s_x[HALF + src] - s_trend[src];            // res recomputed
        }
        s_pp[p * 8 + sg] = acc;
    }
    __syncthreads();
    if (tid < PHASES) {
        float a = 0.0f;
        #pragma unroll
        for (int j = 0; j < 8; ++j) a += s_pp[tid * 8 + j];
        s_season[tid] = a * (1.0f / 342.0f);
    }
    __syncthreads();

    // ---- Stage 5: fused coalesced non-temporal output of all three tensors ----
    const size_t N = (size_t)NROWS * L_LEN;
    float* o_res = out + (size_t)row * L_LEN;      // out_residual
    float* o_tr  = o_res + N;                      // trend
    float* o_se  = o_res + 2 * N;                  // out_seasonal
    for (int j = tid * 4; j < L_LEN; j += TPB * 4) {
        f4 tr = *(const f4*)&s_trend[j];
        f4 xx = *(const f4*)&s_x[HALF + j];
        f4 se, rr;
        #pragma unroll
        for (int e = 0; e < 4; ++e) {
            float sv = s_season[(j + e) % PHASES];
            float r  = xx[e] - tr[e];
            se[e] = sv;
            rr[e] = r - sv;
        }
        __builtin_nontemporal_store(rr, (f4*)(o_res + j));
        __builtin_nontemporal_store(tr, (f4*)(o_tr  + j));
        __builtin_nontemporal_store(se, (f4*)(o_se  + j));
    }
}

extern "C" void kernel_launch(void* const* d_in, const int* in_sizes, int n_in,
                              void* d_out, int out_size, void* d_ws, size_t ws_size,
                              hipStream_t stream)
{
    (void)in_sizes; (void)n_in; (void)d_ws; (void)ws_size; (void)out_size;
    const float* x = (const float*)d_in[0];
    float* out = (float*)d_out;
    decomp_kernel<<<dim3(NROWS), dim3(TPB), 0, stream>>>(x, out);
}